// BasicSparseDeconvolutionBlock_31190052503632
// MI455X (gfx1250) — compile-verified
//
#include <hip/hip_runtime.h>

typedef __attribute__((ext_vector_type(2))) float v2f;
typedef __attribute__((ext_vector_type(4))) float v4f;
typedef __attribute__((ext_vector_type(8))) float v8f;

// ---------------------------------------------------------------------------
// Vectorized zero-fill (128-bit stores); grid-stride.
// ---------------------------------------------------------------------------
__global__ void zero_f32v4(v4f* __restrict__ p, long n4) {
    long i = (long)blockIdx.x * blockDim.x + threadIdx.x;
    const long stride = (long)gridDim.x * blockDim.x;
    const v4f z = {0.0f, 0.0f, 0.0f, 0.0f};
    for (; i < n4; i += stride) p[i] = z;
}

// ---------------------------------------------------------------------------
// Sparse transposed conv: one wave32 per (offset k = blockIdx.y,
// tile of 16 pairs = blockIdx.x).
// A = feats[in_idx[k, base..base+15]] * mask  (16x64 fp32)
// B = W[k]                                    (64x64 fp32)
// D = A @ B via 16 K-steps of V_WMMA_F32_16X16X4_F32 over 4 column tiles,
// then scatter-add rows into out[out_idx] with f32 global atomics.
//
// fp32 A 16x4 layout: lane L holds row M=L%16; VGPR v holds K = v + 2*(L/16).
// fp32 B 4x16 layout: lane L holds col N=L%16; VGPR v holds K = v + 2*(L/16).
// fp32 D 16x16 layout: VGPR j, lane L -> row j + 8*(L/16), col L%16.
// ---------------------------------------------------------------------------
__global__ void __launch_bounds__(32)
spdeconv_wmma(const float* __restrict__ feats,   // [N,64]
              const float* __restrict__ W,       // [K,64,64]
              const float* __restrict__ mask,    // [K,N]
              const int*   __restrict__ in_idx,  // [K,N]
              const int*   __restrict__ out_idx, // [K,N]
              float*       __restrict__ out,     // [N,64] accumulator
              int N)
{
    const int lane = threadIdx.x;
    const int half = lane >> 4;          // 0: K pair base +0, 1: +2
    const int r    = lane & 15;          // row-in-tile (A) / column (B,D)
    const int k    = blockIdx.y;
    const int base = blockIdx.x * 16;
    const long kN  = (long)k * (long)N;

    // --- gather A tile (clamped reads, zero mask for OOB rows; uniform EXEC) ---
    const int   row  = base + r;
    const int   rowc = (row < N) ? row : (N - 1);
    const float mrow = (row < N) ? mask[kN + rowc] : 0.0f;
    const int   gin  = in_idx[kN + rowc];

    v2f a[16];
    const v2f* frow = (const v2f*)(feats + (long)gin * 64 + half * 2);
    #pragma unroll
    for (int t = 0; t < 16; ++t) {
        v2f av = frow[2 * t];            // feats[gin][4t + 2*half .. +1]
        av.x *= mrow;
        av.y *= mrow;
        a[t] = av;
    }

    // --- WMMA accumulation: 4 column tiles of 16, K = 64 in steps of 4 ---
    const float* Wk = W + (long)k * 4096;
    v8f c[4] = {};
    #pragma unroll
    for (int t = 0; t < 16; ++t) {
        const int k0 = 4 * t + half * 2;
        const float* b0 = Wk + (long)k0 * 64 + r;
        #pragma unroll
        for (int ct = 0; ct < 4; ++ct) {
            v2f b;
            b.x = b0[ct * 16];           // B[k0    ][ct*16 + r]
            b.y = b0[ct * 16 + 64];      // B[k0 + 1][ct*16 + r]
            c[ct] = __builtin_amdgcn_wmma_f32_16x16x4_f32(
                false, a[t], false, b, (short)0, c[ct], false, false);
        }
    }

    // --- scatter-add D rows; masked-out rows are exactly zero -> skip atomics ---
    #pragma unroll
    for (int j = 0; j < 8; ++j) {
        const int orow = base + j + 8 * half;
        if (orow < N) {
            const float mj = mask[kN + orow];
            if (mj != 0.0f) {
                const int go = out_idx[kN + orow];
                float* op = out + (long)go * 64 + r;
                atomicAdd(op +  0, c[0][j]);
                atomicAdd(op + 16, c[1][j]);
                atomicAdd(op + 32, c[2][j]);
                atomicAdd(op + 48, c[3][j]);
            }
        }
    }
}

// ---------------------------------------------------------------------------
// Per-channel sum / sum-of-squares over N rows (64 channels), 128-bit loads.
// Each thread owns 4 consecutive channels; 16 threads cover one row.
// Thread-local accumulation, 8 atomics per thread at the end.
// ---------------------------------------------------------------------------
__global__ void bn_stats(const v4f* __restrict__ out4, float* __restrict__ stats, int N) {
    const int ch4     = threadIdx.x & 15;            // channel group (4 ch each)
    const int rowsPB  = blockDim.x >> 4;
    int row           = blockIdx.x * rowsPB + (threadIdx.x >> 4);
    const int rstride = gridDim.x * rowsPB;
    v4f s  = {0.0f, 0.0f, 0.0f, 0.0f};
    v4f s2 = {0.0f, 0.0f, 0.0f, 0.0f};
    for (; row < N; row += rstride) {
        const v4f v = out4[(long)row * 16 + ch4];
        s  += v;
        s2 += v * v;
    }
    const int ch = ch4 * 4;
    #pragma unroll
    for (int e = 0; e < 4; ++e) {
        atomicAdd(&stats[ch + e],      s[e]);
        atomicAdd(&stats[64 + ch + e], s2[e]);
    }
}

// ---------------------------------------------------------------------------
// y = relu((x - mean) * rsqrt(var + eps) * gamma + beta), biased var (/N).
// 128-bit load/store per thread.
// ---------------------------------------------------------------------------
__global__ void bn_relu(v4f* __restrict__ out4, const float* __restrict__ stats,
                        const float* __restrict__ gamma, const float* __restrict__ beta,
                        int N) {
    const long total4 = (long)N * 16;
    long i = (long)blockIdx.x * blockDim.x + threadIdx.x;
    if (i >= total4) return;
    const int ch = (int)(i & 15) * 4;
    const float invN = 1.0f / (float)N;
    v4f v = out4[i];
    #pragma unroll
    for (int e = 0; e < 4; ++e) {
        const float mean = stats[ch + e] * invN;
        const float var  = stats[64 + ch + e] * invN - mean * mean;
        const float inv  = rsqrtf(var + 1e-5f);
        const float y    = (v[e] - mean) * inv * gamma[ch + e] + beta[ch + e];
        v[e] = fmaxf(y, 0.0f);
    }
    out4[i] = v;
}

// ---------------------------------------------------------------------------
// Inputs (setup_inputs order): feats[N,64] f32, W[K,64,64] f32, gamma[64] f32,
// beta[64] f32, pair_mask[K,N] f32, in_idx[K,N] i32, out_idx[K,N] i32.
// Output: [N,64] f32.
// ---------------------------------------------------------------------------
extern "C" void kernel_launch(void* const* d_in, const int* in_sizes, int n_in,
                              void* d_out, int out_size, void* d_ws, size_t ws_size,
                              hipStream_t stream) {
    const float* feats     = (const float*)d_in[0];
    const float* W         = (const float*)d_in[1];
    const float* gamma     = (const float*)d_in[2];
    const float* beta      = (const float*)d_in[3];
    const float* pair_mask = (const float*)d_in[4];
    const int*   in_idx    = (const int*)d_in[5];
    const int*   out_idx   = (const int*)d_in[6];
    float* out   = (float*)d_out;
    float* stats = (float*)d_ws;          // 128 floats: [sum(64) | sumsq(64)]

    const int  N     = in_sizes[0] / 64;
    const int  Koff  = in_sizes[4] / N;   // 27
    const long total = (long)N * 64;

    // 1) zero accumulator + stats (128-bit stores)
    zero_f32v4<<<2048, 256, 0, stream>>>((v4f*)out, total / 4);
    zero_f32v4<<<1, 32, 0, stream>>>((v4f*)stats, 32);

    // 2) gather -> WMMA matmul -> atomic scatter; 2D grid kills the div/rem
    dim3 grid((N + 15) / 16, Koff);
    spdeconv_wmma<<<grid, 32, 0, stream>>>(
        feats, W, pair_mask, in_idx, out_idx, out, N);

    // 3) batch stats (128-bit loads)
    bn_stats<<<2048, 256, 0, stream>>>((const v4f*)out, stats, N);

    // 4) normalize + ReLU in place (128-bit load/store)
    const long total4 = total / 4;
    bn_relu<<<(int)((total4 + 255) / 256), 256, 0, stream>>>(
        (v4f*)out, stats, gamma, beta, N);
}